// LocallyConnected2d_34926674051578
// MI455X (gfx1250) — compile-verified
//
#include <hip/hip_runtime.h>
#include <cstddef>
#include <cstdint>

typedef float v2f __attribute__((ext_vector_type(2)));
typedef float v8f __attribute__((ext_vector_type(8)));

namespace {
constexpr int BATCH = 32;
constexpr int CIN   = 128;
constexpr int H     = 96, W = 96;
constexpr int KH    = 3,  KW = 3;
constexpr int HOUT  = 32, WOUT = 32;
constexpr int COUT  = 128;
constexpr int KDIM  = CIN * KH * KW;   // 1152
constexpr int KC    = 64;              // K-chunk staged in LDS
constexpr int NCHUNK = KDIM / KC;      // 18
constexpr int LSTR  = KC + 4;          // 68: stride for A[m][k] and B[o][k];
                                       // 68 % 64 == 4 -> frag b64 reads hit all 64 banks once
constexpr int NTHREADS = 512;          // 16 wave32
}

// Low 32 bits of the flat address of a __shared__ object == LDS byte offset.
__device__ __forceinline__ unsigned lds_off(const void* p) {
  return (unsigned)(uintptr_t)p;
}

// CDNA5 async copies global -> LDS (ASYNCcnt); no builtin coverage -> inline asm.
__device__ __forceinline__ void async_ld_b128_nt(unsigned lds, const float* g) {
  asm volatile("global_load_async_to_lds_b128 %0, %1, off th:TH_LOAD_NT"
               :: "v"(lds), "v"(g) : "memory");
}
__device__ __forceinline__ void async_ld_b32(unsigned lds, const float* g) {
  asm volatile("global_load_async_to_lds_b32 %0, %1, off"
               :: "v"(lds), "v"(g) : "memory");
}
__device__ __forceinline__ void wait_async0() {
  asm volatile("s_wait_asynccnt 0x0" ::: "memory");
}

__global__ __launch_bounds__(NTHREADS, 1)
void lc2d_wmma_f32(const float* __restrict__ x,
                   const float* __restrict__ weight,
                   const float* __restrict__ bias,
                   float* __restrict__ out)
{
  // double-buffered K-chunks, both row-major along k (async copy can't transpose)
  __shared__ alignas(16) float Al[2][BATCH][LSTR];  // x patch  [m][k]  17.4 KB
  __shared__ alignas(16) float Bl[2][COUT][LSTR];   // weight   [o][k]  69.6 KB

  const int loc = blockIdx.x;            // 0..1023 -> (i, j)
  const int oi  = loc >> 5;
  const int oj  = loc & 31;
  const int tid = threadIdx.x;

  const float* __restrict__ wloc = weight + (size_t)loc * COUT * KDIM;

  // --- per-thread staging assignments ---
  // weight: thread covers one 64B line: o = tid%128, 16 consecutive k (4 x b128)
  const int w_o  = tid & (COUT - 1);
  const int w_kb = (tid >> 7) << 4;      // 0,16,32,48
  const float* wthr = wloc + (size_t)w_o * KDIM + w_kb;
  // x: thread covers m = tid/16, 4 consecutive k (4 x b32 gather)
  const int a_m  = tid >> 4;
  const int a_kb = (tid & 15) << 2;

  auto issue_chunk = [&](int kc, int buf) {
    const float* ws = wthr + kc * KC;
    const unsigned bdst = lds_off(&Bl[buf][w_o][w_kb]);
#pragma unroll
    for (int u = 0; u < 4; ++u)
      async_ld_b128_nt(bdst + 16u * u, ws + 4 * u);
#pragma unroll
    for (int u = 0; u < 4; ++u) {
      const int k  = kc * KC + a_kb + u;     // k = c*9 + p*3 + q
      const int c  = k / 9;
      const int r9 = k - 9 * c;
      const int p  = r9 / 3;
      const int q  = r9 - 3 * p;
      async_ld_b32(lds_off(&Al[buf][a_m][a_kb + u]),
                   x + (((size_t)a_m * CIN + c) * H + (oi * KH + p)) * W
                     + oj * KW + q);
    }
  };

  issue_chunk(0, 0);

  // --- wave tiling: 2 M-tiles x 8 N-tiles of 16x16 ---
  const int lane  = tid & 31;
  const int wv    = tid >> 5;
  const int mt    = wv >> 3;                 // 0..1
  const int nt    = wv & 7;                  // 0..7
  const int rowA  = (mt << 4) + (lane & 15); // batch row for A frag
  const int ocol  = (nt << 4) + (lane & 15); // Cout row for B frag
  const int khalf = (lane >> 4) << 1;        // lanes 0-15: K0,K1; 16-31: K2,K3

  v8f acc = {};

  wait_async0();
  __syncthreads();

  for (int kc = 0; kc < NCHUNK; ++kc) {
    const int buf = kc & 1;
    if (kc + 1 < NCHUNK) issue_chunk(kc + 1, buf ^ 1);  // overlap with WMMA

#pragma unroll
    for (int k = 0; k < KC; k += 4) {
      v2f a = *(const v2f*)&Al[buf][rowA][k + khalf];
      v2f b = *(const v2f*)&Bl[buf][ocol][k + khalf];
      acc = __builtin_amdgcn_wmma_f32_16x16x4_f32(false, a, false, b,
                                                  (short)0, acc, false, false);
    }

    wait_async0();       // my chunk kc+1 async copies have landed in LDS
    __syncthreads();     // everyone's have
  }

  // --- epilogue: bias + NT store ---
  const float bv = bias[(size_t)loc * COUT + ocol];
  const int   mb = (mt << 4) + ((lane >> 4) << 3);  // M = r + 8*(lane>=16)
#pragma unroll
  for (int r = 0; r < 8; ++r) {
    const int b = mb + r;
    const float v = acc[r] + bv;
    __builtin_nontemporal_store(
        v, out + (((size_t)b * COUT + ocol) * HOUT + oi) * WOUT + oj);
  }
}

extern "C" void kernel_launch(void* const* d_in, const int* in_sizes, int n_in,
                              void* d_out, int out_size, void* d_ws, size_t ws_size,
                              hipStream_t stream) {
  const float* x    = (const float*)d_in[0];
  const float* w    = (const float*)d_in[1];
  const float* bias = (const float*)d_in[2];
  float* out        = (float*)d_out;

  dim3 grid(HOUT * WOUT);   // 1024 independent GEMM locations
  dim3 block(NTHREADS);
  hipLaunchKernelGGL(lc2d_wmma_f32, grid, block, 0, stream, x, w, bias, out);
}